// FinancialGNN_4629974745887
// MI455X (gfx1250) — compile-verified
//
#include <hip/hip_runtime.h>
#include <math.h>

#define N_NODES 10000
#define N_EDGES 320000
#define WPR     313          // words per bitmask row: ceil(10000/32)
#define HID     32
#define IN_DIM  6
#define N_TILES (N_NODES / 16)   // 625

typedef __attribute__((ext_vector_type(2))) float v2f;
typedef __attribute__((ext_vector_type(8))) float v8f;

// ---------------------------------------------------------------- utilities
__global__ void zero_u32(unsigned* __restrict__ p, int n) {
    int i = blockIdx.x * blockDim.x + threadIdx.x;
    int stride = gridDim.x * blockDim.x;
    for (; i < n; i += stride) p[i] = 0u;
}

// Scatter edges (set semantics via bit OR) + self loops into bitmask adjacency.
__global__ void scatter_kernel(const int* __restrict__ e0, const int* __restrict__ e1,
                               unsigned* __restrict__ mask) {
    int i = blockIdx.x * blockDim.x + threadIdx.x;
    int total = N_EDGES + N_NODES;
    int stride = gridDim.x * blockDim.x;
    for (; i < total; i += stride) {
        int s, d;
        if (i < N_EDGES) { s = e0[i]; d = e1[i]; }
        else             { s = i - N_EDGES; d = s; }   // self loop (A + I)
        atomicOr(&mask[(size_t)s * WPR + (d >> 5)], 1u << (d & 31));
    }
}

// deg[i] = popcount(row i); dinv = (deg + 1e-8)^-1/2. One wave per row.
__global__ void deg_kernel(const unsigned* __restrict__ mask, float* __restrict__ dinv) {
    int lane = threadIdx.x & 31;
    int wid  = (blockIdx.x * blockDim.x + threadIdx.x) >> 5;
    if (wid >= N_NODES) return;
    const unsigned* row = mask + (size_t)wid * WPR;
    int cnt = 0;
    for (int w = lane; w < WPR; w += 32) cnt += __popc(row[w]);
    for (int off = 16; off > 0; off >>= 1) cnt += __shfl_xor(cnt, off, 32);
    if (lane == 0) dinv[wid] = rsqrtf((float)cnt + 1e-8f);
}

// Layer 0: SpMM over bitmask (6 features) fused with (6x32) GEMM + bias,
// plus in-register BN statistic accumulation.  One wave per row, grid-stride.
__global__ void spmm_gemm0(const unsigned* __restrict__ mask, const float* __restrict__ dinv,
                           const float* __restrict__ x, const float* __restrict__ W0,
                           const float* __restrict__ b0, float* __restrict__ hpre0,
                           float* __restrict__ sum0, float* __restrict__ sq0) {
    int lane = threadIdx.x & 31;
    int wid  = (blockIdx.x * blockDim.x + threadIdx.x) >> 5;
    int nw   = (gridDim.x * blockDim.x) >> 5;
    float w0c[IN_DIM];
#pragma unroll
    for (int k = 0; k < IN_DIM; ++k) w0c[k] = W0[k * HID + lane];
    float bias = b0[lane];
    float ls = 0.f, lq = 0.f;
    for (int row = wid; row < N_NODES; row += nw) {
        const unsigned* mrow = mask + (size_t)row * WPR;
        float acc = 0.f;                    // lanes 0..5 hold feature columns
        for (int w = 0; w < WPR; ++w) {
            unsigned word = mrow[w];
            while (word) {                  // wave-uniform bit scan
                int j = (w << 5) + __builtin_ctz(word);
                word &= word - 1u;
                float dj = dinv[j];
                if (lane < IN_DIM) acc += dj * x[j * IN_DIM + lane];
            }
        }
        acc *= dinv[row];                   // agg = dinv_i * sum_j dinv_j x_j
        float h = bias;                     // h = agg @ W0 + b0 via shuffles
#pragma unroll
        for (int k = 0; k < IN_DIM; ++k) h += __shfl(acc, k, 32) * w0c[k];
        hpre0[row * HID + lane] = h;
        ls += h; lq += h * h;
    }
    atomicAdd(&sum0[lane], ls);
    atomicAdd(&sq0[lane], lq);
}

// Column mean / 1/sqrt(var+eps) from accumulated sums (biased variance).
__global__ void bnstat(const float* __restrict__ sum, const float* __restrict__ sq,
                       float* __restrict__ mu, float* __restrict__ rstd) {
    int c = threadIdx.x;
    float m = sum[c] * (1.0f / (float)N_NODES);
    float v = sq[c] * (1.0f / (float)N_NODES) - m * m;
    mu[c]   = m;
    rstd[c] = rsqrtf(v + 1e-5f);
}

__global__ void bnrelu0(const float* __restrict__ hpre, const float* __restrict__ mu,
                        const float* __restrict__ rstd, const float* __restrict__ g,
                        const float* __restrict__ be, float* __restrict__ h0) {
    int i = blockIdx.x * blockDim.x + threadIdx.x;
    int stride = gridDim.x * blockDim.x;
    for (; i < N_NODES * HID; i += stride) {
        int c = i & (HID - 1);
        float v = (hpre[i] - mu[c]) * rstd[c] * g[c] + be[c];
        h0[i] = fmaxf(v, 0.f);
    }
}

// Layer 1 SpMM: agg2[row, lane] = dinv_i * sum_j dinv_j * h0[j, lane]
__global__ void spmm1(const unsigned* __restrict__ mask, const float* __restrict__ dinv,
                      const float* __restrict__ h0, float* __restrict__ agg2) {
    int lane = threadIdx.x & 31;
    int wid  = (blockIdx.x * blockDim.x + threadIdx.x) >> 5;
    int nw   = (gridDim.x * blockDim.x) >> 5;
    for (int row = wid; row < N_NODES; row += nw) {
        const unsigned* mrow = mask + (size_t)row * WPR;
        float acc = 0.f;
        for (int w = 0; w < WPR; ++w) {
            unsigned word = mrow[w];
            while (word) {
                int j = (w << 5) + __builtin_ctz(word);
                word &= word - 1u;
                acc += dinv[j] * h0[j * HID + lane];
            }
        }
        agg2[row * HID + lane] = acc * dinv[row];
    }
}

// Layer 1 GEMM: hpre2 = agg2(10000x32) @ W1(32x32) + b1 using
// V_WMMA_F32_16X16X4_F32.  Each wave: one 16-row slab, two 16x16 C tiles,
// 8 K-steps -> 16 WMMAs per slab.  B fragments hoisted (W1 uniform).
__global__ void wmma_gemm1(const float* __restrict__ agg2, const float* __restrict__ W1,
                           const float* __restrict__ b1, float* __restrict__ hpre2,
                           float* __restrict__ sum1, float* __restrict__ sq1) {
    int lane = threadIdx.x & 31;
    int wid  = (blockIdx.x * blockDim.x + threadIdx.x) >> 5;
    int nw   = (gridDim.x * blockDim.x) >> 5;
    int m  = lane & 15;
    int hi = lane >> 4;                    // upper half-wave carries K+2/K+3 (A) / rows+8 (D)
    float bias0 = b1[m];
    float bias1 = b1[16 + m];

    v2f bf0[8], bf1[8];                    // B fragments: column N=m per lane
#pragma unroll
    for (int kk = 0; kk < 8; ++kk) {
        int kb = 4 * kk + 2 * hi;
        bf0[kk].x = W1[kb * HID + m];
        bf0[kk].y = W1[(kb + 1) * HID + m];
        bf1[kk].x = W1[kb * HID + 16 + m];
        bf1[kk].y = W1[(kb + 1) * HID + 16 + m];
    }

    float s0 = 0.f, q0 = 0.f, s1 = 0.f, q1 = 0.f;
    for (int tile = wid; tile < N_TILES; tile += nw) {   // wave-uniform: EXEC all-ones
        int row0 = tile * 16;
        v8f c0 = {}; v8f c1 = {};
#pragma unroll
        for (int kk = 0; kk < 8; ++kk) {
            int kb = 4 * kk + 2 * hi;
            v2f a;                          // A fragment: row M=m per lane
            a.x = agg2[(row0 + m) * HID + kb];
            a.y = agg2[(row0 + m) * HID + kb + 1];
            c0 = __builtin_amdgcn_wmma_f32_16x16x4_f32(false, a, false, bf0[kk],
                                                       (short)0, c0, false, false);
            c1 = __builtin_amdgcn_wmma_f32_16x16x4_f32(false, a, false, bf1[kk],
                                                       (short)0, c1, false, false);
        }
#pragma unroll
        for (int r = 0; r < 8; ++r) {      // D layout: VGPR r -> row r (+8 upper half)
            int row = row0 + r + 8 * hi;
            float v0 = c0[r] + bias0;
            float v1 = c1[r] + bias1;
            hpre2[row * HID + m]      = v0;
            hpre2[row * HID + 16 + m] = v1;
            s0 += v0; q0 += v0 * v0;
            s1 += v1; q1 += v1 * v1;
        }
    }
    atomicAdd(&sum1[m],      s0);
    atomicAdd(&sq1[m],       q0);
    atomicAdd(&sum1[16 + m], s1);
    atomicAdd(&sq1[16 + m],  q1);
}

// BN1 + ReLU + residual + attention/logits heads, fused.  One thread per node.
__global__ void final_kernel(const float* __restrict__ hpre2, const float* __restrict__ h0,
                             const float* __restrict__ mu, const float* __restrict__ rstd,
                             const float* __restrict__ g1, const float* __restrict__ be1,
                             const float* __restrict__ Wo, const float* __restrict__ bo,
                             const float* __restrict__ Wa, const float* __restrict__ ba,
                             float* __restrict__ out) {
    int i = blockIdx.x * blockDim.x + threadIdx.x;
    if (i >= N_NODES) return;
    float l0 = bo[0], l1 = bo[1], za = ba[0];
#pragma unroll
    for (int c = 0; c < HID; ++c) {
        float v = (hpre2[i * HID + c] - mu[c]) * rstd[c] * g1[c] + be1[c];
        v = fmaxf(v, 0.f);
        float h = h0[i * HID + c] + v;     // residual
        l0 += h * Wo[c * 2 + 0];
        l1 += h * Wo[c * 2 + 1];
        za += h * Wa[c];
    }
    out[2 * i]     = l0;
    out[2 * i + 1] = l1;
    out[2 * N_NODES + i] = 1.f / (1.f + __expf(-za));
}

// ---------------------------------------------------------------- launcher
extern "C" void kernel_launch(void* const* d_in, const int* in_sizes, int n_in,
                              void* d_out, int out_size, void* d_ws, size_t ws_size,
                              hipStream_t stream) {
    const float* x   = (const float*)d_in[0];
    const int*   ei  = (const int*)d_in[1];
    const int*   e0  = ei;
    const int*   e1  = ei + N_EDGES;
    const float* W0  = (const float*)d_in[2];
    const float* b0  = (const float*)d_in[3];
    const float* g0  = (const float*)d_in[4];
    const float* be0 = (const float*)d_in[5];
    const float* W1  = (const float*)d_in[6];
    const float* b1  = (const float*)d_in[7];
    const float* g1  = (const float*)d_in[8];
    const float* be1 = (const float*)d_in[9];
    const float* Wo  = (const float*)d_in[10];
    const float* bo  = (const float*)d_in[11];
    const float* Wa  = (const float*)d_in[12];
    const float* ba  = (const float*)d_in[13];
    float* out = (float*)d_out;

    // workspace carving (256B aligned)
    char* ws = (char*)d_ws;
    size_t off = 0;
    auto carve = [&](size_t bytes) -> void* {
        void* p = ws + off;
        off += (bytes + 255) & ~(size_t)255;
        return p;
    };
    unsigned* mask  = (unsigned*)carve((size_t)N_NODES * WPR * sizeof(unsigned)); // 12.5 MB
    float*    dinv  = (float*)carve(N_NODES * sizeof(float));
    float*    stats = (float*)carve(8 * 32 * sizeof(float));
    float* sum0 = stats +   0; float* sq0  = stats +  32;
    float* mu0  = stats +  64; float* rstd0= stats +  96;
    float* sum1 = stats + 128; float* sq1  = stats + 160;
    float* mu1  = stats + 192; float* rstd1= stats + 224;
    float* hpre0 = (float*)carve((size_t)N_NODES * HID * sizeof(float));
    float* h0    = (float*)carve((size_t)N_NODES * HID * sizeof(float));
    float* agg2  = (float*)carve((size_t)N_NODES * HID * sizeof(float));
    float* hpre2 = (float*)carve((size_t)N_NODES * HID * sizeof(float));

    // 1) clear bitmask + stat accumulators
    zero_u32<<<512, 256, 0, stream>>>(mask, N_NODES * WPR);
    zero_u32<<<1, 256, 0, stream>>>((unsigned*)stats, 8 * 32);
    // 2) build set-semantics adjacency (+ self loops)
    scatter_kernel<<<(N_EDGES + N_NODES + 255) / 256, 256, 0, stream>>>(e0, e1, mask);
    // 3) degrees -> dinv
    deg_kernel<<<(N_NODES * 32 + 255) / 256, 256, 0, stream>>>(mask, dinv);
    // 4) layer0: SpMM + GEMM(6x32) + BN stat accumulation
    spmm_gemm0<<<128, 256, 0, stream>>>(mask, dinv, x, W0, b0, hpre0, sum0, sq0);
    // 5) BN0 stats, then BN+ReLU
    bnstat<<<1, 32, 0, stream>>>(sum0, sq0, mu0, rstd0);
    bnrelu0<<<(N_NODES * HID + 255) / 256, 256, 0, stream>>>(hpre0, mu0, rstd0, g0, be0, h0);
    // 6) layer1: SpMM then WMMA f32 GEMM(32x32) + BN stat accumulation
    spmm1<<<128, 256, 0, stream>>>(mask, dinv, h0, agg2);
    wmma_gemm1<<<80, 256, 0, stream>>>(agg2, W1, b1, hpre2, sum1, sq1);
    // 7) BN1 stats, then fused BN+ReLU+residual+heads
    bnstat<<<1, 32, 0, stream>>>(sum1, sq1, mu1, rstd1);
    final_kernel<<<(N_NODES + 255) / 256, 256, 0, stream>>>(hpre2, h0, mu1, rstd1,
                                                            g1, be1, Wo, bo, Wa, ba, out);
}